// GraphConvolution_39402029973789
// MI455X (gfx1250) — compile-verified
//
#include <hip/hip_runtime.h>

typedef __attribute__((ext_vector_type(2))) float v2f;
typedef __attribute__((ext_vector_type(8))) float v8f;

#define D          128   // D_IN == D_OUT
#define LDS_STRIDE 132   // 128 + 4 pad -> conflict-free strided LDS reads

// ---------------------------------------------------------------------------
// Kernel 1: support = X @ W using V_WMMA_F32_16X16X4_F32 (fp32-exact path).
// Grid: Mtot/16 blocks of 256 threads (8 waves). Wave w computes the 16x16
// tile at columns [16w, 16w+16); the 16x128 row-stripe of X is staged in LDS.
// ---------------------------------------------------------------------------
__global__ __launch_bounds__(256)
void gcn_gemm_wmma_f32(const float* __restrict__ A,   // [Mtot, 128]
                       const float* __restrict__ W,   // [128, 128]
                       float* __restrict__ S)         // [Mtot, 128]
{
    __shared__ float As[16 * LDS_STRIDE];

    const int row0  = blockIdx.x << 4;
    const int tid   = threadIdx.x;
    const int wave  = tid >> 5;
    const int lane  = tid & 31;
    const int mrow  = lane & 15;
    const int half  = lane >> 4;       // 0: lanes 0-15, 1: lanes 16-31
    const int ncol0 = wave << 4;

    // Cooperative, coalesced stage of the 16x128 A stripe.
    for (int idx = tid; idx < 16 * D; idx += 256) {
        const int r = idx >> 7;
        const int c = idx & (D - 1);
        As[r * LDS_STRIDE + c] = A[(long long)(row0 + r) * D + c];
    }
    __syncthreads();

    v8f acc = {};
#pragma unroll
    for (int k0 = 0; k0 < D; k0 += 4) {
        const int k = k0 + 2 * half;   // ISA f32 A/B frag: half-wave picks K pair
        v2f a, b;
        a.x = As[mrow * LDS_STRIDE + k + 0];          // A[M=mrow][K=k]
        a.y = As[mrow * LDS_STRIDE + k + 1];          // A[M=mrow][K=k+1]
        b.x = W[(k + 0) * D + ncol0 + mrow];          // B[K=k  ][N=mrow]
        b.y = W[(k + 1) * D + ncol0 + mrow];          // B[K=k+1][N=mrow]
        acc = __builtin_amdgcn_wmma_f32_16x16x4_f32(
            /*neg_a=*/false, a, /*neg_b=*/false, b,
            /*c_mod=*/(short)0, acc, /*reuse_a=*/false, /*reuse_b=*/false);
    }

    // C/D layout: VGPR i -> M = i + 8*half, N = ncol0 + mrow.
#pragma unroll
    for (int i = 0; i < 8; ++i) {
        S[(long long)(row0 + i + 8 * half) * D + ncol0 + mrow] = acc[i];
    }
}

// ---------------------------------------------------------------------------
// Kernel 2: out[b,n,f] = bias[f]   (accumulator base for the scatter-add)
// ---------------------------------------------------------------------------
__global__ __launch_bounds__(256)
void gcn_init_bias(float* __restrict__ out, const float* __restrict__ bias,
                   long long total)
{
    const long long i = (long long)blockIdx.x * blockDim.x + threadIdx.x;
    if (i < total) out[i] = bias[i & (D - 1)];
}

// ---------------------------------------------------------------------------
// Kernel 3: one wave per (batch, edge). Lane l handles features [4l, 4l+4):
// gather support[b, col] (coalesced 512B/wave, L2-resident), scale by the
// edge weight, scatter-add into out[b, row] with global fp32 atomics.
// ---------------------------------------------------------------------------
__global__ __launch_bounds__(256)
void gcn_edge_scatter(const float* __restrict__ S,
                      const float* __restrict__ vals,
                      const int*   __restrict__ rows,
                      const int*   __restrict__ cols,
                      float* __restrict__ out, int E, int N)
{
    const int gw   = blockIdx.x * 8 + (threadIdx.x >> 5);   // global wave id
    const int lane = threadIdx.x & 31;
    if (gw >= 4 * E) return;                                // B = 4
    const int b = gw / E;
    const int e = gw - b * E;

    const float v = vals[e];
    const long long srcOff = ((long long)b * N + cols[e]) * D;
    const long long dstOff = ((long long)b * N + rows[e]) * D;

    const float4 m = ((const float4*)(S + srcOff))[lane];
    float* dst = out + dstOff + lane * 4;
    atomicAdd(dst + 0, v * m.x);
    atomicAdd(dst + 1, v * m.y);
    atomicAdd(dst + 2, v * m.z);
    atomicAdd(dst + 3, v * m.w);
}

// ---------------------------------------------------------------------------
// Kernel 4: ReLU in place (float4).
// ---------------------------------------------------------------------------
__global__ __launch_bounds__(256)
void gcn_relu(float4* __restrict__ out, long long total4)
{
    const long long i = (long long)blockIdx.x * blockDim.x + threadIdx.x;
    if (i < total4) {
        float4 x = out[i];
        x.x = fmaxf(x.x, 0.0f);
        x.y = fmaxf(x.y, 0.0f);
        x.z = fmaxf(x.z, 0.0f);
        x.w = fmaxf(x.w, 0.0f);
        out[i] = x;
    }
}

extern "C" void kernel_launch(void* const* d_in, const int* in_sizes, int n_in,
                              void* d_out, int out_size, void* d_ws, size_t ws_size,
                              hipStream_t stream) {
    const float* x    = (const float*)d_in[0];   // [B, N, 128] fp32
    const float* w    = (const float*)d_in[1];   // [128, 128]  fp32
    const float* bias = (const float*)d_in[2];   // [128]       fp32
    const float* vals = (const float*)d_in[3];   // [E]         fp32
    const int*   rows = (const int*)  d_in[4];   // [E]         i32
    const int*   cols = (const int*)  d_in[5];   // [E]         i32
    float* out     = (float*)d_out;              // [B, N, 128] fp32
    float* support = (float*)d_ws;               // [B, N, 128] fp32 scratch

    const int B = 4, N = 50000;
    const int E = in_sizes[3];
    const long long Mtot  = (long long)B * N;    // 200000 rows (÷16 exactly)
    const long long total = Mtot * D;            // 25.6M output elements

    // 1) support = X @ W  (WMMA fp32)
    gcn_gemm_wmma_f32<<<(int)(Mtot / 16), 256, 0, stream>>>(x, w, support);

    // 2) out = broadcast(bias)
    gcn_init_bias<<<(int)((total + 255) / 256), 256, 0, stream>>>(out, bias, total);

    // 3) scatter-add edge messages (one wave per batch-edge)
    const int nwaves = B * E;
    gcn_edge_scatter<<<(nwaves + 7) / 8, 256, 0, stream>>>(support, vals, rows,
                                                           cols, out, E, N);

    // 4) ReLU
    const long long total4 = total / 4;
    gcn_relu<<<(int)((total4 + 255) / 256), 256, 0, stream>>>((float4*)out, total4);
}